// GeodesicPathIntegrator_87462714016195
// MI455X (gfx1250) — compile-verified
//
#include <hip/hip_runtime.h>
#include <hip/hip_bf16.h>
#include <math.h>

// Geometry (fixed by the reference):
//   trace: (33, 8192, 256) f32 = (T, B, 64 quats * 4); steps = 32
#define T_STEPS      32
#define B_ROWS       8192
#define ROW_BYTES    (8192u * 256u * 4u)    // 8 MiB per time-row
#define CB_ROWS      16                      // batch rows per workgroup chunk
#define CHUNK_BYTES  (CB_ROWS * 256u * 4u)   // 16 KiB per (time-row, chunk)
#define CHUNK_FLOATS (CB_ROWS * 256u)        // 4096 floats
#define NUM_WG       (B_ROWS / CB_ROWS)      // 512 b-chunks
#define T_SPLIT      2                       // time-axis split (16 steps each)
#define STEPS_HALF   (T_STEPS / T_SPLIT)     // 16

struct C2 { float r, i; };
__device__ __forceinline__ C2 cmul(float xr, float xi, float yr, float yi) {
    // (xr + i xi) * (yr + i yi), same structure as complex mult in the einsum
    C2 o; o.r = xr * yr - xi * yi; o.i = xr * yi + xi * yr; return o;
}
__device__ __forceinline__ C2 cadd(C2 p, C2 q) {
    C2 o; o.r = p.r + q.r; o.i = p.i + q.i; return o;
}

__global__ __launch_bounds__(32) void geo_zero_kernel(float* acc) {
    acc[threadIdx.x] = 0.0f;
}

__global__ __launch_bounds__(256) void geo_main_kernel(const float* __restrict__ trace,
                                                       float* __restrict__ acc) {
    // 4-deep ring of 16 KiB LDS buffers: rows t, t+1 resident, t+2 and t+3 in flight.
    __shared__ __align__(16) float smem[4 * CHUNK_FLOATS];

    const unsigned tid  = threadIdx.x;            // 0..255
    const unsigned wg   = blockIdx.x;             // 0..511  (b-chunk)
    const unsigned half = blockIdx.y;             // 0..1    (time half)
    const unsigned r0   = half * STEPS_HALF;      // first time-row of this half
    const unsigned long long gbase = (unsigned long long)trace;
    const unsigned chunk_goff = wg * CHUNK_BYTES;

    // Issue one time-row chunk (16 KiB) as 4 async B128 transfers per thread
    // (= 4 ASYNCcnt increments per wave).
    auto issue_row = [&](unsigned r, unsigned buf) {
        unsigned g0 = r * ROW_BYTES + chunk_goff + tid * 16u;
        unsigned l0 = buf * CHUNK_BYTES + tid * 16u;
#pragma unroll
        for (int j = 0; j < 4; ++j) {
            unsigned goff = g0 + (unsigned)j * 4096u;
            unsigned loff = l0 + (unsigned)j * 4096u;
            // vdst = LDS byte address, vaddr = u32 byte offset, saddr = 64-bit base
            asm volatile("global_load_async_to_lds_b128 %0, %1, %2"
                         :: "v"(loff), "v"(goff), "s"(gbase)
                         : "memory");
        }
    };

    issue_row(r0 + 0u, 0u);
    issue_row(r0 + 1u, 1u);
    issue_row(r0 + 2u, 2u);

    const float inv_pi = 0.31830988618379067154f;

    for (unsigned s = 0; s < STEPS_HALF; ++s) {
        // Prefetch row t+3 two steps ahead; rows t+2/t+3 may stay in flight.
        if (s + 3u <= (unsigned)STEPS_HALF) {
            issue_row(r0 + s + 3u, (s + 3u) & 3u);
            asm volatile("s_wait_asynccnt 0x8" ::: "memory");   // t+2,t+3 pending
        } else if (s + 2u <= (unsigned)STEPS_HALF) {
            asm volatile("s_wait_asynccnt 0x4" ::: "memory");   // only t+2 pending
        } else {
            asm volatile("s_wait_asynccnt 0x0" ::: "memory");   // drain
        }
        __syncthreads();

        const float4* __restrict__ q0p =
            reinterpret_cast<const float4*>(&smem[(s & 3u) * CHUNK_FLOATS]);
        const float4* __restrict__ q1p =
            reinterpret_cast<const float4*>(&smem[((s + 1u) & 3u) * CHUNK_FLOATS]);

        float local = 0.0f;
#pragma unroll
        for (int i = 0; i < 4; ++i) {
            unsigned e = tid + (unsigned)i * 256u;   // 0..1023
            float4 q0 = q0p[e];   // s_t      : (a, b, c, d)
            float4 q1 = q1p[e];   // s_{t+1}  : (A, B, C, D)
            float a = q0.x, b = q0.y, c = q0.z, d = q0.w;
            float A = q1.x, Bq = q1.y, Cq = q1.z, D = q1.w;

            // s = [[a+id, b+ic], [-b+ic, a-id]];  S[i][l] = sum_j s1[i][j]*conj(s0[l][j])
            C2 S00 = cadd(cmul( A,   D,  a, -d), cmul( Bq,  Cq,  b, -c));
            C2 S01 = cadd(cmul( A,   D, -b, -c), cmul( Bq,  Cq,  a,  d));
            C2 S10 = cadd(cmul(-Bq,  Cq, a, -d), cmul( A,  -D,   b, -c));
            C2 S11 = cadd(cmul(-Bq,  Cq, -b, -c), cmul( A,  -D,  a,  d));

            // det = S00*S11 - S01*S10
            float detRe = (S00.r * S11.r - S00.i * S11.i) - (S01.r * S10.r - S01.i * S10.i);
            float detIm = (S00.r * S11.i + S00.i * S11.r) - (S01.r * S10.i + S01.i * S10.r);

            float eta = atan2f(detIm, detRe) * inv_pi;
            local += fabsf(eta);
        }

        // wave32 tree reduction, then one f32 atomic per wave per step
#pragma unroll
        for (int off = 16; off > 0; off >>= 1)
            local += __shfl_down(local, off, 32);
        if ((tid & 31u) == 0u)
            atomicAdd(&acc[r0 + s], local);

        __syncthreads();   // buffer (s&3) may be re-issued next iteration
    }
}

__global__ __launch_bounds__(32) void geo_final_kernel(const float* __restrict__ acc,
                                                       float* __restrict__ out) {
    unsigned t = threadIdx.x;                       // 32 lanes, one per step
    float v = acc[t] * (1.0f / 524288.0f);          // step_eta[t] = mean(|eta|)
#pragma unroll
    for (int off = 16; off > 0; off >>= 1)
        v += __shfl_down(v, off, 32);
    if (t == 0) {
        out[0] = v * (1.0f / 32.0f);                // avg_curvature
        out[1] = v;                                 // total_eta
    }
}

extern "C" void kernel_launch(void* const* d_in, const int* in_sizes, int n_in,
                              void* d_out, int out_size, void* d_ws, size_t ws_size,
                              hipStream_t stream) {
    (void)in_sizes; (void)n_in; (void)out_size; (void)ws_size;
    const float* trace = (const float*)d_in[0];
    float* acc = (float*)d_ws;       // 32 per-step accumulators
    float* out = (float*)d_out;      // [avg_curvature, total_eta]

    geo_zero_kernel<<<1, 32, 0, stream>>>(acc);
    geo_main_kernel<<<dim3(NUM_WG, T_SPLIT), 256, 0, stream>>>(trace, acc);
    geo_final_kernel<<<1, 32, 0, stream>>>(acc, out);
}